// LabelDependencyGCN_62955630624884
// MI455X (gfx1250) — compile-verified
//
#include <hip/hip_runtime.h>

typedef float v2f __attribute__((ext_vector_type(2)));
typedef float v4f __attribute__((ext_vector_type(4)));
typedef float v8f __attribute__((ext_vector_type(8)));

#define N_NODES   100000
#define NUM_LABELS 128
#define HIDDEN      64
#define N_EDGES  3200000
#define E_TOT    (N_EDGES + N_NODES)   // edges + self loops

// ---------------------------------------------------------------------------
// degree / normalization
// ---------------------------------------------------------------------------
__global__ void gcn_init_deg(unsigned* __restrict__ deg) {
    int i = blockIdx.x * blockDim.x + threadIdx.x;
    if (i < N_NODES) deg[i] = 1u;            // self loop contributes 1
}

__global__ void gcn_accum_deg(const long long* __restrict__ ei,
                              unsigned* __restrict__ deg) {
    int e = blockIdx.x * blockDim.x + threadIdx.x;
    if (e < N_EDGES) {
        int dst = (int)ei[N_EDGES + e];
        atomicAdd(&deg[dst], 1u);
    }
}

__global__ void gcn_dinv(const unsigned* __restrict__ deg,
                         float* __restrict__ dinv) {
    int i = blockIdx.x * blockDim.x + threadIdx.x;
    if (i < N_NODES) dinv[i] = rsqrtf((float)deg[i]);
}

__global__ void gcn_zero_f32(float* __restrict__ p, int n) {
    int i = blockIdx.x * blockDim.x + threadIdx.x;
    if (i < n) p[i] = 0.0f;
}

// ---------------------------------------------------------------------------
// edge scatter: agg[dst] += dinv[src]*dinv[dst] * feat[src]
// one wave32 per edge, NF/32 features per lane (vector loads, f32 atomics)
// ---------------------------------------------------------------------------
template <int NF>
__global__ void gcn_scatter(const float* __restrict__ feat,
                            const long long* __restrict__ ei,
                            const float* __restrict__ dinv,
                            float* __restrict__ agg) {
    const int lane = threadIdx.x & 31;
    const long long warp =
        ((long long)blockIdx.x * blockDim.x + threadIdx.x) >> 5;
    if (warp >= (long long)E_TOT) return;

    int src, dst;
    if (warp < (long long)N_EDGES) {
        src = (int)ei[warp];
        dst = (int)ei[N_EDGES + warp];
    } else {
        src = dst = (int)(warp - N_EDGES);   // self loop
    }
    const float norm = dinv[src] * dinv[dst];

    constexpr int FP = NF / 32;              // features per lane (2 or 4)
    const float* frow = feat + (long long)src * NF + lane * FP;
    float*       arow = agg  + (long long)dst * NF + lane * FP;

    if constexpr (FP == 4) {
        v4f v = *(const v4f*)frow;
        unsafeAtomicAdd(&arow[0], v.x * norm);
        unsafeAtomicAdd(&arow[1], v.y * norm);
        unsafeAtomicAdd(&arow[2], v.z * norm);
        unsafeAtomicAdd(&arow[3], v.w * norm);
    } else {
        v2f v = *(const v2f*)frow;
        unsafeAtomicAdd(&arow[0], v.x * norm);
        unsafeAtomicAdd(&arow[1], v.y * norm);
    }
}

// ---------------------------------------------------------------------------
// WMMA f32 GEMM: out[M,NF] = A[M,K] @ B[K,NF] + bias (optional relu)
// One wave per 16-row strip; NT = NF/16 accumulator tiles; K-step 4 using
// V_WMMA_F32_16X16X4_F32 (exact f32, matches reference precision).
//
// VGPR layouts per CDNA5 ISA 7.12.2:
//   A 16x4 : lane l (h=l>>4, m=l&15): vgpr v -> A[m][k0 + 2h + v]
//   B 4x16 : lane l (n=l&15):         vgpr v -> B[k0 + 2h + v][n]
//   C/D    : lane l:                  vgpr v -> C[v + 8h][n]
// ---------------------------------------------------------------------------
template <int K, int NF, bool RELU>
__global__ void gcn_wmma_gemm(const float* __restrict__ A,
                              const float* __restrict__ B,
                              const float* __restrict__ bias,
                              float* __restrict__ out) {
    constexpr int NT = NF / 16;
    const int lane = threadIdx.x;            // blockDim == 32
    const int h    = lane >> 4;
    const int l16  = lane & 15;
    const int m0   = blockIdx.x * 16;

    v8f acc[NT];
#pragma unroll
    for (int t = 0; t < NT; ++t) {
        float bv = bias[t * 16 + l16];       // bias depends only on column
#pragma unroll
        for (int v = 0; v < 8; ++v) acc[t][v] = bv;
    }

    const float* arow = A + (long long)(m0 + l16) * K + 2 * h;

    for (int k0 = 0; k0 < K; k0 += 4) {
        v2f a = *(const v2f*)(arow + k0);    // A[m][k0+2h], A[m][k0+2h+1]
        const float* bptr = B + (long long)(k0 + 2 * h) * NF + l16;
#pragma unroll
        for (int t = 0; t < NT; ++t) {
            v2f b;
            b.x = bptr[t * 16];              // B[k0+2h  ][n]
            b.y = bptr[NF + t * 16];         // B[k0+2h+1][n]
            acc[t] = __builtin_amdgcn_wmma_f32_16x16x4_f32(
                false, a, false, b, (short)0, acc[t], false, false);
        }
    }

#pragma unroll
    for (int t = 0; t < NT; ++t) {
#pragma unroll
        for (int v = 0; v < 8; ++v) {
            int row = m0 + v + 8 * h;
            float val = acc[t][v];
            if (RELU) val = fmaxf(val, 0.0f);
            out[(long long)row * NF + t * 16 + l16] = val;
        }
    }
}

// ---------------------------------------------------------------------------
// launch
// ---------------------------------------------------------------------------
extern "C" void kernel_launch(void* const* d_in, const int* in_sizes, int n_in,
                              void* d_out, int out_size, void* d_ws, size_t ws_size,
                              hipStream_t stream) {
    const float*     logits = (const float*)d_in[0];
    const long long* ei     = (const long long*)d_in[1];
    const float*     W1     = (const float*)d_in[2];
    const float*     b1     = (const float*)d_in[3];
    const float*     W2     = (const float*)d_in[4];
    const float*     b2     = (const float*)d_in[5];
    float*           out    = (float*)d_out;

    // workspace layout (bytes)
    char* ws = (char*)d_ws;
    unsigned* deg  = (unsigned*)(ws);                              // 400 KB
    float*    dinv = (float*)(ws + (512u << 10));                  // 400 KB
    float*    agg  = (float*)(ws + (1u << 20));                    // 51.2 MB (layer1: N x 128; reused layer2: N x 64)
    float*    x    = (float*)(ws + (1u << 20) + 52u * (1u << 20)); // 25.6 MB

    const int TB = 256;
    const int nodeBlocks = (N_NODES + TB - 1) / TB;
    const int edgeBlocks = (N_EDGES + TB - 1) / TB;
    // one wave32 per edge (incl. self loops) -> 8 edges per 256-thread block
    const long long scatterBlocks = ((long long)E_TOT * 32 + TB - 1) / TB;

    // normalization: deg -> d^{-1/2}
    gcn_init_deg<<<nodeBlocks, TB, 0, stream>>>(deg);
    gcn_accum_deg<<<edgeBlocks, TB, 0, stream>>>(ei, deg);
    gcn_dinv<<<nodeBlocks, TB, 0, stream>>>(deg, dinv);

    // ---- layer 1: agg1 = scatter(logits); x = relu(agg1 @ W1 + b1) ----
    {
        const int n = N_NODES * NUM_LABELS;
        gcn_zero_f32<<<(n + TB - 1) / TB, TB, 0, stream>>>(agg, n);
    }
    gcn_scatter<NUM_LABELS><<<(unsigned)scatterBlocks, TB, 0, stream>>>(
        logits, ei, dinv, agg);
    gcn_wmma_gemm<NUM_LABELS, HIDDEN, true><<<N_NODES / 16, 32, 0, stream>>>(
        agg, W1, b1, x);

    // ---- layer 2: agg2 = scatter(x); out = agg2 @ W2 + b2 ----
    {
        const int n = N_NODES * HIDDEN;
        gcn_zero_f32<<<(n + TB - 1) / TB, TB, 0, stream>>>(agg, n);
    }
    gcn_scatter<HIDDEN><<<(unsigned)scatterBlocks, TB, 0, stream>>>(
        x, ei, dinv, agg);
    gcn_wmma_gemm<HIDDEN, NUM_LABELS, false><<<N_NODES / 16, 32, 0, stream>>>(
        agg, W2, b2, out);
}